// B_188978561578
// MI455X (gfx1250) — compile-verified
//
#include <hip/hip_runtime.h>

// ---------------------------------------------------------------------------
// Reference: x = x / 10 applied y (=8) times, elementwise over 134M f32.
// 512 MiB in + 512 MiB out = 1 GiB HBM traffic -> ~46 us at 23.3 TB/s.
// Memory-bound; WMMA inapplicable. CDNA5 path used instead:
//   * double-buffered GLOBAL_LOAD_ASYNC_TO_LDS_B128 (ASYNCcnt) pipeline
//   * s_wait_asynccnt ping-pong + s_wait_dscnt WAR fence
//   * non-temporal hints (stream is 5x the 192MB L2)
//   * Markstein-refined constant division: 3 VALU ops per correctly-rounded
//     divide (24/elem) instead of the ~9-op v_div_scale sequence.
// Round-1 fix: y is uniform; unswitch the division loop once at kernel entry
// so the y==8 hot path is straight-line VALU with no SALU loop control.
// ---------------------------------------------------------------------------

typedef float f4 __attribute__((ext_vector_type(4)));

#define TPB 256

// Correctly-rounded (normal-range) division by 10.0f:
// r = RN(1/10); y0 = RN(x*r); e = fma(-10,y0,x); y1 = fma(e,r,y0).
__device__ __forceinline__ float div10(float x) {
    const float r = 0.1f;
    float y = x * r;
    float e = __builtin_fmaf(-10.0f, y, x);
    return __builtin_fmaf(e, r, y);
}

__device__ __forceinline__ f4 div10v(f4 v) {
    f4 o;
    o.x = div10(v.x);
    o.y = div10(v.y);
    o.z = div10(v.z);
    o.w = div10(v.w);
    return o;
}

template <bool FIXED8>
__device__ __forceinline__ void stream_body(const f4* __restrict__ in,
                                            f4* __restrict__ out,
                                            int ydiv, unsigned n4,
                                            unsigned numTiles,
                                            f4 (&buf)[2][TPB]) {
    const unsigned tid    = threadIdx.x;
    const unsigned stride = gridDim.x;

    // Generic LDS pointer low 32 bits == byte offset within LDS allocation.
    const unsigned ldsOff0 = (unsigned)(uintptr_t)(&buf[0][tid]);
    const unsigned ldsOff1 = (unsigned)(uintptr_t)(&buf[1][tid]);

    unsigned t = blockIdx.x;
    if (t >= numTiles) return;

    // Prime the pipeline: async-load tile t into buffer 0.
    {
        unsigned idx = t * TPB + tid;
        if (idx >= n4) idx = n4 - 1;            // clamp: keep EXEC uniform
        const f4* src = in + idx;
        asm volatile("global_load_async_to_lds_b128 %0, %1, off th:TH_LOAD_NT"
                     :: "v"(ldsOff0), "v"(src) : "memory");
    }

    unsigned p = 0;
    for (; t < numTiles; t += stride) {
        const unsigned nt = t + stride;
        if (nt < numTiles) {
            unsigned idx = nt * TPB + tid;
            if (idx >= n4) idx = n4 - 1;
            const f4* src = in + idx;
            const unsigned dstOff = (p == 0) ? ldsOff1 : ldsOff0;
            // Prior ds_load of this buffer must complete before the async
            // DMA overwrites it (LDS WAR; DS and async are unordered).
            asm volatile("s_wait_dscnt 0x0" ::: "memory");
            asm volatile("global_load_async_to_lds_b128 %0, %1, off th:TH_LOAD_NT"
                         :: "v"(dstOff), "v"(src) : "memory");
            // 2 async ops outstanding; wait until only the newest remains.
            asm volatile("s_wait_asynccnt 0x1" ::: "memory");
        } else {
            asm volatile("s_wait_asynccnt 0x0" ::: "memory");
        }

        // Consume tile t from LDS (ds_load_b128), compute, NT-store.
        f4 v = buf[p][tid];
        if constexpr (FIXED8) {
#pragma unroll
            for (int i = 0; i < 8; ++i) v = div10v(v);
        } else {
            for (int i = 0; i < ydiv; ++i) v = div10v(v);
        }

        const unsigned oidx = t * TPB + tid;
        if (oidx < n4) __builtin_nontemporal_store(v, out + oidx);

        p ^= 1u;
    }
}

__global__ __launch_bounds__(TPB) void div10_stream_kernel(
        const f4* __restrict__ in, f4* __restrict__ out,
        const int* __restrict__ yptr, unsigned n4, unsigned numTiles) {
    // Per-lane private 16B slots, two buffers for double buffering.
    // No cross-lane / cross-wave sharing -> no workgroup barriers needed.
    __shared__ f4 buf[2][TPB];

    const int ydiv = *yptr;          // uniform scalar (s_load)
    if (ydiv == 8) {
        stream_body<true>(in, out, 8, n4, numTiles, buf);   // hot path
    } else {
        stream_body<false>(in, out, ydiv, n4, numTiles, buf);
    }
}

// Scalar tail for n % 4 != 0 (not hit for this shape, kept for generality).
__global__ void div10_tail_kernel(const float* __restrict__ in,
                                  float* __restrict__ out,
                                  const int* __restrict__ yptr,
                                  unsigned start, unsigned n) {
    unsigned i = start + blockIdx.x * blockDim.x + threadIdx.x;
    if (i < n) {
        float v = in[i];
        int k = *yptr;
        for (int j = 0; j < k; ++j) v = div10(v);
        out[i] = v;
    }
}

extern "C" void kernel_launch(void* const* d_in, const int* in_sizes, int n_in,
                              void* d_out, int out_size, void* d_ws, size_t ws_size,
                              hipStream_t stream) {
    const float* x    = (const float*)d_in[0];
    const int*   yptr = (const int*)d_in[1];
    float*       out  = (float*)d_out;

    const unsigned n  = (unsigned)in_sizes[0];   // 134,217,728
    const unsigned n4 = n / 4u;                  // float4 count

    if (n4 > 0) {
        const unsigned numTiles = (n4 + TPB - 1u) / TPB;
        unsigned grid = numTiles < 16384u ? numTiles : 16384u;
        div10_stream_kernel<<<grid, TPB, 0, stream>>>(
            (const f4*)x, (f4*)out, yptr, n4, numTiles);
    }

    const unsigned rem = n - n4 * 4u;
    if (rem) {
        div10_tail_kernel<<<(rem + 63u) / 64u, 64, 0, stream>>>(
            x, out, yptr, n4 * 4u, n);
    }
}